// GRUSlowStateExtractor_40003325394978
// MI455X (gfx1250) — compile-verified
//
#include <hip/hip_runtime.h>
#include <hip/hip_bf16.h>

// GRUSlowStateExtractor for MI455X (gfx1250), fp32 WMMA 16x16x4.
// B=256, T=1024, IN=7, H=64, gates G=3H=192.
// Workspace: 2 ping-pong activation buffers [T,B,64] f32 = 128 MB total.
// Latency-critical recurrent scan uses dual WMMA accumulator chains per wave.

#define B_   256
#define T_   1024
#define H_   64
#define G_   192
#define EPS_ 1e-8f

typedef float v2f __attribute__((ext_vector_type(2)));
typedef float v4f __attribute__((ext_vector_type(4)));
typedef float v8f __attribute__((ext_vector_type(8)));

__device__ __forceinline__ v8f wmma4(v2f a, v2f b, v8f c) {
  // D(16x16,f32) = A(16x4,f32) * B(4x16,f32) + C
  return __builtin_amdgcn_wmma_f32_16x16x4_f32(
      /*neg_a=*/false, a, /*neg_b=*/false, b,
      /*c_mod=*/(short)0, c, /*reuse_a=*/false, /*reuse_b=*/false);
}

__device__ __forceinline__ float sig_(float x) {
  return 1.0f / (1.0f + __expf(-x));
}
__device__ __forceinline__ float tanh_(float x) {
  x = fminf(fmaxf(x, -15.0f), 15.0f);
  float e = __expf(2.0f * x);
  return (e - 1.0f) / (e + 1.0f);
}

// ---------------------------------------------------------------------------
// Recurrent GRU layer kernel. One workgroup owns 16 batch rows and scans all
// T steps. 12 waves: wave = chunk(0=r,1=z,2=n)*4 + coltile(0..3 of H=64).
// Per step each wave does K=64 (h-side) and (layers 1/2) K=64 (input-side)
// f32 WMMA accumulation into a 16x16 tile using TWO independent WMMA chains
// (even/odd k-blocks) to double ILP on the latency-critical path. Weight
// B-fragments live in VGPRs for the whole kernel; A-fragments come from LDS.
// ---------------------------------------------------------------------------
template <bool IS_L0>
__global__ __launch_bounds__(384) void gru_layer_kernel(
    const float* __restrict__ in,    // IS_L0: x [B,T,7]   else: z_prev [T,B,64]
    const float* __restrict__ h0,    // [B,64] slice for this layer
    const float* __restrict__ W_ih,  // [192, IN]
    const float* __restrict__ W_hh,  // [192, 64]
    const float* __restrict__ b_ih,  // [192]
    const float* __restrict__ b_hh,  // [192]
    const float* __restrict__ resW,  // [64,7] (L0 only)
    const float* __restrict__ resb,  // [64]   (L0 only)
    float* __restrict__ out,         // [T,B,64]  (= y + residual)
    float* __restrict__ hT)          // [B,64]
{
  __shared__ float h_lds[16 * 64];
  __shared__ float in_lds[16 * 64];
  __shared__ float x_lds[16][8];
  __shared__ float r_buf[16 * 64];
  __shared__ float z_buf[16 * 64];

  const int tid   = threadIdx.x;
  const int wave  = __builtin_amdgcn_readfirstlane(tid >> 5);
  const int chunk = __builtin_amdgcn_readfirstlane(wave >> 2);  // 0=r 1=z 2=n
  const int wtile = __builtin_amdgcn_readfirstlane(wave & 3);
  const int lane  = tid & 31;
  const int lrow  = lane & 15;       // M row index within lane half
  const int hi    = lane >> 4;       // 0: K+0/K+1 half, 1: K+2/K+3 half
  const int sel   = hi * 2;
  const int colH  = wtile * 16 + lrow;        // column within H
  const int colG  = chunk * 64 + colH;        // column within 3H gates
  const int bb    = blockIdx.x * 16;          // batch tile base

  // ---- persistent weight B-fragments (registers) -------------------------
  v2f wfh[16];
#pragma unroll
  for (int kb = 0; kb < 16; ++kb)
    wfh[kb] = *(const v2f*)(W_hh + (size_t)colG * H_ + kb * 4 + sel);

  v2f wfi[16];
  if constexpr (!IS_L0) {
#pragma unroll
    for (int kb = 0; kb < 16; ++kb)
      wfi[kb] = *(const v2f*)(W_ih + (size_t)colG * H_ + kb * 4 + sel);
  }

  float wih0r[7], resw7[7], resb0 = 0.0f;
  if constexpr (IS_L0) {
#pragma unroll
    for (int k = 0; k < 7; ++k) wih0r[k] = W_ih[colG * 7 + k];
    if (chunk == 2) {
#pragma unroll
      for (int k = 0; k < 7; ++k) resw7[k] = resW[colH * 7 + k];
      resb0 = resb[colH];
    }
  }

  float bsum = 0.0f, bi = 0.0f, bh = 0.0f;
  if (chunk < 2) {
    bsum = b_ih[colG] + b_hh[colG];
  } else {
    bi = b_ih[colG];
    bh = b_hh[colG];
  }

  // ---- init h in LDS -----------------------------------------------------
  for (int idx = tid; idx < 16 * 64; idx += 384)
    h_lds[idx] = h0[(size_t)(bb + (idx >> 6)) * H_ + (idx & 63)];

  // ---- prefetch input tile for t=0 (layers 1/2) --------------------------
  v4f pre0 = {0.f, 0.f, 0.f, 0.f}, pre1 = {0.f, 0.f, 0.f, 0.f};
  if constexpr (!IS_L0) {
    if (wave < 4) {
      const int row = tid >> 3, c0 = (tid & 7) * 8;
      const float* p = in + ((size_t)0 * B_ + bb + row) * H_ + c0;
      pre0 = *(const v4f*)p;
      pre1 = *(const v4f*)(p + 4);
    }
  }

  const v8f vzero = {0.f, 0.f, 0.f, 0.f, 0.f, 0.f, 0.f, 0.f};

  for (int t = 0; t < T_; ++t) {
    __syncthreads();  // prev iter fully done: h_lds updated, in_lds reads done

    if constexpr (!IS_L0) {
      if (wave < 4) {
        const int row = tid >> 3, c0 = (tid & 7) * 8;
        *(v4f*)&in_lds[row * 64 + c0]     = pre0;
        *(v4f*)&in_lds[row * 64 + c0 + 4] = pre1;
        if (t + 1 < T_) {
          const float* p = in + ((size_t)(t + 1) * B_ + bb + row) * H_ + c0;
          pre0 = *(const v4f*)p;
          pre1 = *(const v4f*)(p + 4);
        }
      }
    } else {
      if (tid < 16) {
        const float* xr = in + (size_t)(bb + tid) * (T_ * 7) + (size_t)t * 7;
#pragma unroll
        for (int k = 0; k < 7; ++k) x_lds[tid][k] = xr[k];
        x_lds[tid][7] = 0.0f;
      }
    }
    __syncthreads();  // input tile staged; h_lds(t) valid

    // ---- matrix core: dual independent WMMA chains per accumulator -------
    v8f accA = vzero, accB = vzero;    // gh (r/z: gh+gi fused)
    v8f ac2A = vzero, ac2B = vzero;    // gi (n-chunk only, layers 1/2)
#pragma unroll
    for (int kb = 0; kb < 16; kb += 2) {
      v2f ah0 = *(const v2f*)&h_lds[lrow * 64 + kb * 4 + sel];
      v2f ah1 = *(const v2f*)&h_lds[lrow * 64 + (kb + 1) * 4 + sel];
      accA = wmma4(ah0, wfh[kb], accA);
      accB = wmma4(ah1, wfh[kb + 1], accB);
    }
    if constexpr (!IS_L0) {
      if (chunk < 2) {
#pragma unroll
        for (int kb = 0; kb < 16; kb += 2) {
          v2f ai0 = *(const v2f*)&in_lds[lrow * 64 + kb * 4 + sel];
          v2f ai1 = *(const v2f*)&in_lds[lrow * 64 + (kb + 1) * 4 + sel];
          accA = wmma4(ai0, wfi[kb], accA);
          accB = wmma4(ai1, wfi[kb + 1], accB);
        }
      } else {
#pragma unroll
        for (int kb = 0; kb < 16; kb += 2) {
          v2f ai0 = *(const v2f*)&in_lds[lrow * 64 + kb * 4 + sel];
          v2f ai1 = *(const v2f*)&in_lds[lrow * 64 + (kb + 1) * 4 + sel];
          ac2A = wmma4(ai0, wfi[kb], ac2A);
          ac2B = wmma4(ai1, wfi[kb + 1], ac2B);
        }
      }
    }
    float acc[8], acc2[8];
#pragma unroll
    for (int i = 0; i < 8; ++i) {
      acc[i]  = accA[i] + accB[i];
      acc2[i] = ac2A[i] + ac2B[i];
    }

    // ---- layer0 scalar input-side (K=7) + residual projection ------------
    float giv[8], resv[8];
    if constexpr (IS_L0) {
#pragma unroll
      for (int i = 0; i < 8; ++i) {
        const int M = i + 8 * hi;
        float s = 0.0f;
#pragma unroll
        for (int k = 0; k < 7; ++k) s += x_lds[M][k] * wih0r[k];
        giv[i] = s;
        if (chunk == 2) {
          float r2 = resb0;
#pragma unroll
          for (int k = 0; k < 7; ++k) r2 += x_lds[M][k] * resw7[k];
          resv[i] = r2;
        }
      }
    }

    // ---- gates -----------------------------------------------------------
    if (chunk == 0) {
#pragma unroll
      for (int i = 0; i < 8; ++i) {
        const int M = i + 8 * hi;
        float g = acc[i] + bsum;
        if constexpr (IS_L0) g += giv[i];
        r_buf[M * 64 + colH] = sig_(g);
      }
    } else if (chunk == 1) {
#pragma unroll
      for (int i = 0; i < 8; ++i) {
        const int M = i + 8 * hi;
        float g = acc[i] + bsum;
        if constexpr (IS_L0) g += giv[i];
        z_buf[M * 64 + colH] = sig_(g);
      }
    }
    __syncthreads();  // r/z published

    if (chunk == 2) {
#pragma unroll
      for (int i = 0; i < 8; ++i) {
        const int M  = i + 8 * hi;
        const float rv  = r_buf[M * 64 + colH];
        const float zv  = z_buf[M * 64 + colH];
        const float hv  = h_lds[M * 64 + colH];
        const float ghn = acc[i] + bh;
        float gin;
        if constexpr (IS_L0) gin = bi + giv[i];
        else                 gin = acc2[i] + bi;
        const float nv = tanh_(gin + rv * ghn);
        const float hn = (1.0f - zv) * nv + zv * hv;
        h_lds[M * 64 + colH] = hn;
        float resid;
        if constexpr (IS_L0) resid = resv[i];
        else                 resid = in_lds[M * 64 + colH];
        out[((size_t)t * B_ + bb + M) * H_ + colH] = hn + resid;
        if (t == T_ - 1) hT[(size_t)(bb + M) * H_ + colH] = hn;
      }
    }
  }
}

// ---------------------------------------------------------------------------
// MLP heads + quaternion post-processing.
// 4 waves / WG, one 16-row tile of the flattened [T*B, 64] activation.
// Stage 1: hidden = relu(z2 @ W1^T + b1) via f32 WMMA for both heads.
// Stage 2: skinny 64->4 and 64->3 dots. Stage 3: quaternion math per row.
// ---------------------------------------------------------------------------
__global__ __launch_bounds__(128) void head_kernel(
    const float* __restrict__ z2,   // [T,B,64]
    const float* __restrict__ x,    // [B,T,7]
    const float* __restrict__ mean_, const float* __restrict__ std_,
    const float* __restrict__ qW1, const float* __restrict__ qb1,
    const float* __restrict__ qW2, const float* __restrict__ qb2,
    const float* __restrict__ wW1, const float* __restrict__ wb1,
    const float* __restrict__ wW2, const float* __restrict__ wb2,
    float* __restrict__ out_qraw, float* __restrict__ out_q,
    float* __restrict__ out_w)
{
  __shared__ float hidq[16 * 64];
  __shared__ float hidw[16 * 64];
  __shared__ float dq[16 * 4];

  const int tid  = threadIdx.x;
  const int wave = __builtin_amdgcn_readfirstlane(tid >> 5);
  const int lane = tid & 31;
  const int lrow = lane & 15;
  const int hi   = lane >> 4;
  const int sel  = hi * 2;
  const int colg = wave * 16 + lrow;
  const size_t rowbase = (size_t)blockIdx.x * 16;

  v2f qf[16], wf[16];
#pragma unroll
  for (int kb = 0; kb < 16; ++kb) {
    qf[kb] = *(const v2f*)(qW1 + (size_t)colg * 64 + kb * 4 + sel);
    wf[kb] = *(const v2f*)(wW1 + (size_t)colg * 64 + kb * 4 + sel);
  }
  const float qb = qb1[colg], wb = wb1[colg];

  v8f aq = {0.f, 0.f, 0.f, 0.f, 0.f, 0.f, 0.f, 0.f};
  v8f aw = {0.f, 0.f, 0.f, 0.f, 0.f, 0.f, 0.f, 0.f};
#pragma unroll
  for (int kb = 0; kb < 16; ++kb) {
    v2f a = *(const v2f*)(z2 + (rowbase + lrow) * 64 + kb * 4 + sel);
    aq = wmma4(a, qf[kb], aq);
    aw = wmma4(a, wf[kb], aw);
  }
#pragma unroll
  for (int i = 0; i < 8; ++i) {
    const int M = i + 8 * hi;
    hidq[M * 64 + colg] = fmaxf(aq[i] + qb, 0.0f);
    hidw[M * 64 + colg] = fmaxf(aw[i] + wb, 0.0f);
  }
  __syncthreads();

  if (tid < 112) {
    const int row = tid / 7, oc = tid % 7;
    const size_t rg = rowbase + row;      // rg = t*B + b
    const int t = (int)(rg >> 8), b = (int)(rg & 255);
    if (oc < 4) {
      float s = qb2[oc];
      for (int k = 0; k < 64; ++k) s += hidq[row * 64 + k] * qW2[oc * 64 + k];
      dq[row * 4 + oc] = s;
    } else {
      const int o2 = oc - 4;
      float s = wb2[o2];
      for (int k = 0; k < 64; ++k) s += hidw[row * 64 + k] * wW2[o2 * 64 + k];
      out_w[((size_t)b * T_ + t) * 3 + o2] = s;
    }
  }
  __syncthreads();

  if (tid < 16) {
    const size_t rg = rowbase + tid;
    const int t = (int)(rg >> 8), b = (int)(rg & 255);
    const float* xr = x + ((size_t)b * T_ + t) * 7;
    float q[4];
    float nn = 0.0f;
#pragma unroll
    for (int j = 0; j < 4; ++j) {
      q[j] = xr[j] * std_[j] + mean_[j];
      nn += q[j] * q[j];
    }
    const float inv = 1.0f / (sqrtf(nn) + EPS_);
    const float s0  = (q[0] < 0.0f) ? -1.0f : 1.0f;  // sign(q_in[0]) after >0 scale
    float raw[4];
    float n2 = 0.0f;
#pragma unroll
    for (int j = 0; j < 4; ++j) {
      raw[j] = q[j] * inv * s0 + dq[tid * 4 + j];
      n2 += raw[j] * raw[j];
    }
    const float inv2 = 1.0f / (sqrtf(n2) + EPS_);
    const float s1   = (raw[0] < 0.0f) ? -1.0f : 1.0f;
    const size_t ob  = ((size_t)b * T_ + t) * 4;
#pragma unroll
    for (int j = 0; j < 4; ++j) {
      out_qraw[ob + j] = raw[j];
      out_q[ob + j]    = raw[j] * inv2 * s1;
    }
  }
}

extern "C" void kernel_launch(void* const* d_in, const int* in_sizes, int n_in,
                              void* d_out, int out_size, void* d_ws, size_t ws_size,
                              hipStream_t stream) {
  const float* x     = (const float*)d_in[0];
  const float* h0    = (const float*)d_in[1];
  const float* mean_ = (const float*)d_in[2];
  const float* std_  = (const float*)d_in[3];
  const float* Wih0  = (const float*)d_in[4];
  const float* Whh0  = (const float*)d_in[5];
  const float* bih0  = (const float*)d_in[6];
  const float* bhh0  = (const float*)d_in[7];
  const float* Wih1  = (const float*)d_in[8];
  const float* Whh1  = (const float*)d_in[9];
  const float* bih1  = (const float*)d_in[10];
  const float* bhh1  = (const float*)d_in[11];
  const float* Wih2  = (const float*)d_in[12];
  const float* Whh2  = (const float*)d_in[13];
  const float* bih2  = (const float*)d_in[14];
  const float* bhh2  = (const float*)d_in[15];
  const float* resW  = (const float*)d_in[16];
  const float* resb  = (const float*)d_in[17];
  const float* qW1   = (const float*)d_in[18];
  const float* qb1   = (const float*)d_in[19];
  const float* qW2   = (const float*)d_in[20];
  const float* qb2   = (const float*)d_in[21];
  const float* wW1   = (const float*)d_in[22];
  const float* wb1   = (const float*)d_in[23];
  const float* wW2   = (const float*)d_in[24];
  const float* wb2   = (const float*)d_in[25];

  float* buf0 = (float*)d_ws;                       // z0, then z2  [T,B,64]
  float* buf1 = buf0 + (size_t)T_ * B_ * H_;        // z1           [T,B,64]

  float* o        = (float*)d_out;
  float* out_qraw = o;                                  // [B,T,4]
  float* out_q    = o + (size_t)B_ * T_ * 4;            // [B,T,4]
  float* out_w    = o + (size_t)B_ * T_ * 8;            // [B,T,3]
  float* hT       = o + (size_t)B_ * T_ * 8 + (size_t)B_ * T_ * 3;  // [3,B,64]

  dim3 gridR(B_ / 16), blockR(384);
  gru_layer_kernel<true><<<gridR, blockR, 0, stream>>>(
      x, h0, Wih0, Whh0, bih0, bhh0, resW, resb, buf0, hT);
  gru_layer_kernel<false><<<gridR, blockR, 0, stream>>>(
      buf0, h0 + (size_t)B_ * H_, Wih1, Whh1, bih1, bhh1, nullptr, nullptr,
      buf1, hT + (size_t)B_ * H_);
  gru_layer_kernel<false><<<gridR, blockR, 0, stream>>>(
      buf1, h0 + (size_t)2 * B_ * H_, Wih2, Whh2, bih2, bhh2, nullptr, nullptr,
      buf0, hT + (size_t)2 * B_ * H_);

  dim3 gridH((T_ * B_) / 16), blockH(128);
  head_kernel<<<gridH, blockH, 0, stream>>>(
      buf0, x, mean_, std_, qW1, qb1, qW2, qb2, wW1, wb1, wW2, wb2,
      out_qraw, out_q, out_w);
}